// VectorizationLoss_55344948576612
// MI455X (gfx1250) — compile-verified
//
#include <hip/hip_runtime.h>
#include <hip/hip_bf16.h>

typedef float v2f __attribute__((ext_vector_type(2)));
typedef float v8f __attribute__((ext_vector_type(8)));

#define NPTS 2048
#define BATCH 8
#define SBUF (BATCH * NPTS)          // elements per state buffer
#define L2E 1.4426950408889634f      // log2(e)
#define LN2 0.6931471805599453f

// One softmin update for all 4 Sinkhorn potentials (blockIdx.z = task):
//   task 0: f_ab  (rows=x, cols=y, h = hconst + hscale*g_ab_old)
//   task 1: g_ab  (rows=y, cols=x, h = hconst + hscale*f_ab_old)
//   task 2: f_aa  (rows=x, cols=x, h = hconst + hscale*f_aa_old)
//   task 3: g_bb  (rows=y, cols=y, h = hconst + hscale*g_bb_old)
// out = alpha*old + beta*(-eps*logsumexp_j(h_j - C_ij/eps)),  C = 0.5|xi-yj|^2
//
// The full argument (log2 domain) is computed inside V_WMMA_F32_16X16X4_F32:
//   A[i][:] = (ie2*xi.x, ie2*xi.y, -rr_i(-m_i), 1),  B[:][j] = (yj.x, yj.y, 1, q_j)
//   D[i][j] = ie2*<xi,yj> - rr_i + q_j  (= arg, or arg - m_i in pass 2)
__global__ __launch_bounds__(256) void sinkhorn_iter_kernel(
    const float* __restrict__ xpts, const float* __restrict__ ypts,
    const float* __restrict__ st_old, float* __restrict__ st_new,
    float eps, float inv_eps, float hscale, float hconst,
    float alpha, float beta)
{
  __shared__ float2 lcomb[2 * NPTS]; // [0,NPTS): (y.x,y.y)  [NPTS,2N): (1, q)  32 KB
  __shared__ float  lm[8 * 16];      // per-wave row-max transpose scratch

  const int task = blockIdx.z;
  const int b    = blockIdx.y;
  const int tid  = threadIdx.x;
  const int lane = tid & 31;
  const int wave = tid >> 5;

  const float* Xp = (task == 0 || task == 2) ? xpts : ypts;
  const float* Yp = (task == 0 || task == 3) ? ypts : xpts;
  const int hidx  = (task == 0) ? 1 : (task == 1) ? 0 : task;
  const float* hsrc = st_old + hidx * SBUF + b * NPTS;
  const float* oldv = st_old + task * SBUF + b * NPTS;
  float*       outv = st_new + task * SBUF + b * NPTS;

  const float ie2 = inv_eps * L2E; // 1/eps in log2 units

  // ---- stage columns into LDS ----
  const float2* Y2 = (const float2*)Yp;
  for (int j = tid; j < NPTS; j += 256) {
    float2 yp = Y2[b * NPTS + j];
    float h = hconst;
    if (hscale != 0.0f) h += hscale * hsrc[j];
    float sy = 0.5f * (yp.x * yp.x + yp.y * yp.y);
    lcomb[j] = yp;                                      // K=0,1 rows of B
    lcomb[NPTS + j] = make_float2(1.0f, (h - sy * inv_eps) * L2E); // K=2,3
  }
  __syncthreads();

  // ---- row tile: A matrix (16x4 f32) ----
  const int i0 = (blockIdx.x * 8 + wave) * 16;
  const float2* X2 = (const float2*)Xp;
  float2 xp = X2[b * NPTS + i0 + (lane & 15)];
  float nx2 = 0.5f * (xp.x * xp.x + xp.y * xp.y);  // sx of row i0+(lane&15)
  const bool lo = (lane < 16);
  const int rbase = lo ? 0 : 8;

  v2f amat;                        // lanes 0-15: K=0,1 ; lanes 16-31: K=2,3
  amat.x = lo ? (ie2 * xp.x) : (-nx2 * ie2);   // K0 = ie2*x.x | K2 = -rr_i
  amat.y = lo ? (ie2 * xp.y) : 1.0f;           // K1 = ie2*x.y | K3 = 1

  const int colbase = (lo ? 0 : NPTS) + (lane & 15);

  float m[8];
#pragma unroll
  for (int r = 0; r < 8; ++r) m[r] = -3.4e38f;

  // ---- pass 1: per-lane running max of arg (straight out of WMMA) ----
  for (int jt = 0; jt < NPTS / 16; ++jt) {
    float2 cv = lcomb[colbase + jt * 16];
    v2f bmat; bmat.x = cv.x; bmat.y = cv.y;
    v8f c = {};
    c = __builtin_amdgcn_wmma_f32_16x16x4_f32(false, amat, false, bmat,
                                              (short)0, c, false, false);
#pragma unroll
    for (int r = 0; r < 8; ++r) m[r] = fmaxf(m[r], c[r]);
  }

  // ---- exact row max: merge across 16-lane half ----
#pragma unroll
  for (int r = 0; r < 8; ++r) {
#pragma unroll
    for (int mask = 8; mask >= 1; mask >>= 1)
      m[r] = fmaxf(m[r], __shfl_xor(m[r], mask, 32));
  }
  // transpose slot->row through LDS (in-order within wave, no barrier needed)
  if ((lane & 15) == 0) {
#pragma unroll
    for (int r = 0; r < 8; ++r) lm[wave * 16 + rbase + r] = m[r];
  }
  float m_row = lm[wave * 16 + (lane & 15)];

  // fold -m_row into A's K=2 entry for pass 2: WMMA emits arg - m directly
  v2f amat2 = amat;
  amat2.x = amat.x - (lo ? 0.0f : m_row);

  float s[8];
#pragma unroll
  for (int r = 0; r < 8; ++r) s[r] = 0.0f;

  // ---- pass 2: sum of 2^(arg - m) ----
  for (int jt = 0; jt < NPTS / 16; ++jt) {
    float2 cv = lcomb[colbase + jt * 16];
    v2f bmat; bmat.x = cv.x; bmat.y = cv.y;
    v8f c = {};
    c = __builtin_amdgcn_wmma_f32_16x16x4_f32(false, amat2, false, bmat,
                                              (short)0, c, false, false);
#pragma unroll
    for (int r = 0; r < 8; ++r) s[r] += __builtin_amdgcn_exp2f(c[r]);
  }

  // ---- all lanes share the same row max -> plain shuffle-add reduction ----
#pragma unroll
  for (int r = 0; r < 8; ++r) {
#pragma unroll
    for (int mask = 8; mask >= 1; mask >>= 1)
      s[r] += __shfl_xor(s[r], mask, 32);
  }

  // ---- epilogue: res = -eps * ln2 * (log2(s) + m); mix with old state ----
  if ((lane & 15) == 0) {
#pragma unroll
    for (int r = 0; r < 8; ++r) {
      int row = i0 + rbase + r;
      float lse2 = __builtin_amdgcn_logf(s[r]) + m[r];
      float res  = -eps * LN2 * lse2;
      float v = beta * res;
      if (alpha != 0.0f) v = fmaf(alpha, oldv[row], v);
      outv[row] = v;
    }
  }
}

// loss[b] = mean_i(f_ab - f_aa) + mean_j(g_ab - g_bb) + mean_b(cp_loss)
__global__ __launch_bounds__(256) void sinkhorn_final_kernel(
    const float* __restrict__ st, const float* __restrict__ target,
    const float* __restrict__ cpts, float* __restrict__ out)
{
  __shared__ float red[8];
  const int b = blockIdx.x;
  const int tid = threadIdx.x;
  const float* f_ab = st + 0 * SBUF + b * NPTS;
  const float* g_ab = st + 1 * SBUF + b * NPTS;
  const float* f_aa = st + 2 * SBUF + b * NPTS;
  const float* g_bb = st + 3 * SBUF + b * NPTS;
  float acc = 0.0f;
  for (int i = tid; i < NPTS; i += 256)
    acc += (f_ab[i] - f_aa[i]) + (g_ab[i] - g_bb[i]);
#pragma unroll
  for (int mask = 16; mask >= 1; mask >>= 1)
    acc += __shfl_xor(acc, mask, 32);
  if ((tid & 31) == 0) red[tid >> 5] = acc;
  __syncthreads();
  if (tid == 0) {
    float tot = 0.0f;
#pragma unroll
    for (int w = 0; w < 8; ++w) tot += red[w];
    float cp = 0.0f;
    for (int bb = 0; bb < BATCH; ++bb) {
      for (int d = 0; d < 2; ++d) {
        // controlpoints (B,32,4,2): [bb,0,0,d] and [bb,0,3,d]
        float c0 = cpts[bb * 256 + d];
        float cL = cpts[bb * 256 + 6 + d];
        float t0 = target[bb * (NPTS * 2) + d];
        float tL = target[bb * (NPTS * 2) + (NPTS - 1) * 2 + d];
        float d0 = c0 - t0, dL = cL - tL;
        cp += 0.5f * (d0 * d0 + dL * dL);   // P=2 -> diff^2/2
      }
    }
    cp /= (float)BATCH;
    out[b] = tot / (float)NPTS + cp;
  }
}

extern "C" void kernel_launch(void* const* d_in, const int* in_sizes, int n_in,
                              void* d_out, int out_size, void* d_ws, size_t ws_size,
                              hipStream_t stream) {
  const float* x    = (const float*)d_in[0];   // out (8,32,64,2) == (8,2048,2)
  const float* y    = (const float*)d_in[1];   // target (8,2048,2)
  const float* cpts = (const float*)d_in[2];   // controlpoints (8,32,4,2)
  float* out = (float*)d_out;
  float* ws  = (float*)d_ws;                   // needs 8*SBUF floats = 512 KB
  float* setA = ws;
  float* setB = ws + 4 * SBUF;

  // eps schedule: 16,4,1,...,0.00390625, then 0.0025 x4  (11 values)
  double eps_list[32]; int ne = 0;
  double e = 16.0;                  // DIAMETER^P
  const double tgt = 0.05 * 0.05;   // BLUR^P
  while (e > tgt && ne < 28) { eps_list[ne++] = e; e *= 0.25; }
  for (int k = 0; k < 4; ++k) eps_list[ne++] = tgt;

  const float hconst = -7.6246189861593985f;   // -ln(2048) = log_a = log_b
  dim3 grid(16, BATCH, 4), block(256);

  // init with eps0: pure softmin of constant h (alpha=0, hscale=0)
  {
    float e0 = (float)eps_list[0];
    sinkhorn_iter_kernel<<<grid, block, 0, stream>>>(
        x, y, setA, setA, e0, 1.0f / e0, 0.0f, hconst, 0.0f, 1.0f);
  }
  // eps-scaling iterations (ping-pong state sets)
  float* cur = setA;
  float* nxt = setB;
  for (int k = 1; k < ne; ++k) {
    float ek = (float)eps_list[k];
    sinkhorn_iter_kernel<<<grid, block, 0, stream>>>(
        x, y, cur, nxt, ek, 1.0f / ek, 1.0f / ek, hconst, 0.5f, 0.5f);
    float* t = cur; cur = nxt; nxt = t;
  }
  sinkhorn_final_kernel<<<BATCH, 256, 0, stream>>>(cur, y, cpts, out);
}